// MultiHeadAttentionWithRelativePosition_44427141710214
// MI455X (gfx1250) — compile-verified
//
#include <hip/hip_runtime.h>
#include <hip/hip_bf16.h>
#include <stddef.h>
#include <stdint.h>

// ---------------------------------------------------------------------------
// MHA with relative position (CDNA5 / gfx1250, wave32).
// Matrix math: V_WMMA_F32_16X16X4_F32 (full fp32 precision path).
// Data movement: global_load_async_to_lds_b128 (ASYNCcnt) double-buffered
// tile staging, so inner loops are ds_load -> v_wmma with DMA in flight.
// ---------------------------------------------------------------------------

typedef float v2f __attribute__((ext_vector_type(2)));
typedef float v8f __attribute__((ext_vector_type(8)));

__device__ __forceinline__ v8f v8f_zero() {
    v8f z;
    #pragma unroll
    for (int i = 0; i < 8; ++i) z[i] = 0.0f;
    return z;
}

__device__ __forceinline__ v8f wmma4(v2f a, v2f b, v8f c) {
    return __builtin_amdgcn_wmma_f32_16x16x4_f32(
        false, a, false, b, (short)0, c, false, false);
}

// Async DMA: copy 16 bytes global -> LDS per lane (one b128 per lane).
// Generic pointers to LDS carry the LDS byte offset in their low 32 bits.
__device__ __forceinline__ void async_copy16(const float* g, const float* l) {
    unsigned lo = (unsigned)(uintptr_t)l;
    asm volatile("global_load_async_to_lds_b128 %0, %1, off"
                 :: "v"(lo), "v"(g) : "memory");
}
__device__ __forceinline__ void wait_async0() {
    asm volatile("s_wait_asynccnt 0" ::: "memory");
}
__device__ __forceinline__ void wait_ds0() {
    asm volatile("s_wait_dscnt 0" ::: "memory");
}

#define B_DIM 2
#define S_DIM 1024
#define D_MODEL 1024
#define NHEAD 16
#define HDIM 64
#define NROWS (B_DIM * S_DIM)

// ---------------------------------------------------------------------------
// Y = X(2048x1024) @ W(1024x1024) + bias.  Block: 4 waves, 64x64 tile.
// Both operands async-staged into double-buffered LDS; DMA for chunk k+1
// is issued before computing chunk k.
// Row strides: Xs 20 floats (80B, 16B-aligned, gcd(20,64)=4 -> 16 distinct
// banks for the 16 A-fragment lanes); Ws 72 floats (288B, 16B-aligned).
// ---------------------------------------------------------------------------
__global__ __launch_bounds__(128)
void gemm_proj(const float* __restrict__ X, const float* __restrict__ W,
               const float* __restrict__ bias, float* __restrict__ Y, int mode) {
    __shared__ float Xs[2][64][20];
    __shared__ float Ws[2][16][72];

    const int tid  = threadIdx.x;
    const int lane = tid & 31;
    const int wv   = tid >> 5;
    const int half = lane >> 4;
    const int lm   = lane & 15;
    const int rbase = blockIdx.x * 64;
    const int cbase = blockIdx.y * 64;

    auto stage = [&](int k0, int bb) {
        #pragma unroll
        for (int i = 0; i < 2; ++i) {                 // X tile: 64x16 floats
            int e4 = tid + i * 128;                   // 256 float4s
            int r = e4 >> 2, c4 = e4 & 3;
            async_copy16(X + (size_t)(rbase + r) * D_MODEL + k0 + c4 * 4,
                         &Xs[bb][r][c4 * 4]);
        }
        #pragma unroll
        for (int i = 0; i < 2; ++i) {                 // W tile: 16x64 floats
            int e4 = tid + i * 128;
            int kr = e4 >> 4, c4 = e4 & 15;
            async_copy16(W + (size_t)(k0 + kr) * D_MODEL + cbase + c4 * 4,
                         &Ws[bb][kr][c4 * 4]);
        }
    };

    v8f acc[4];
    #pragma unroll
    for (int i = 0; i < 4; ++i) acc[i] = v8f_zero();

    stage(0, 0);
    int buf = 0;
    for (int k0 = 0; k0 < D_MODEL; k0 += 16) {
        wait_async0();          // this chunk's DMA landed
        wait_ds0();             // my reads of the other buffer are done
        __syncthreads();        // everyone aligned before overwriting it
        if (k0 + 16 < D_MODEL) stage(k0 + 16, buf ^ 1);

        const float (*Xb)[20] = Xs[buf];
        const float (*Wb)[72] = Ws[buf];
        #pragma unroll
        for (int kk = 0; kk < 16; kk += 4) {
            v2f a;
            a.x = Xb[16 * wv + lm][kk + 2 * half];
            a.y = Xb[16 * wv + lm][kk + 2 * half + 1];
            #pragma unroll
            for (int nf = 0; nf < 4; ++nf) {
                v2f b;
                b.x = Wb[kk + 2 * half][nf * 16 + lm];
                b.y = Wb[kk + 2 * half + 1][nf * 16 + lm];
                acc[nf] = wmma4(a, b, acc[nf]);
            }
        }
        buf ^= 1;
    }

    #pragma unroll
    for (int nf = 0; nf < 4; ++nf) {
        const int c  = cbase + nf * 16 + lm;
        const float bv = bias[c];
        #pragma unroll
        for (int i = 0; i < 8; ++i) {
            const int r = rbase + 16 * wv + 8 * half + i;
            const float val = acc[nf][i] + bv;
            const int bidx = r >> 10, s = r & 1023;
            const int h = c >> 6, d = c & 63;
            if (mode == 1)        // khT: [B,H,D,S]
                Y[(((size_t)bidx * NHEAD + h) * HDIM + d) * S_DIM + s] = val;
            else if (mode == 3)   // plain [rows, D_MODEL]
                Y[(size_t)r * D_MODEL + c] = val;
            else                  // [B,H,S,D]
                Y[(((size_t)bidx * NHEAD + h) * S_DIM + s) * HDIM + d] = val;
        }
    }
}

// ---------------------------------------------------------------------------
// qe[b,h,q,k] = sum_d qh[b,h,q,d] * e_r[k,d], masked by (k >= 1023 - q).
// ---------------------------------------------------------------------------
__global__ __launch_bounds__(128)
void qe_kernel(const float* __restrict__ qh, const float* __restrict__ e_r,
               float* __restrict__ qe) {
    __shared__ float Qs[64][68];
    __shared__ float Es[64][68];   // Es[d][k_local] = e_r[kb+k_local][d]

    const int tid  = threadIdx.x;
    const int lane = tid & 31;
    const int wv   = tid >> 5;
    const int half = lane >> 4;
    const int lm   = lane & 15;
    const int kb = blockIdx.x * 64;
    const int qb = blockIdx.y * 64;
    const int bh = blockIdx.z;
    const float* Q = qh + (size_t)bh * S_DIM * HDIM;

    #pragma unroll
    for (int i = 0; i < 8; ++i) {                  // Q tile 64x64, async b128
        int e4 = tid + i * 128;
        int r = e4 >> 4, c4 = e4 & 15;
        async_copy16(Q + (size_t)(qb + r) * HDIM + c4 * 4, &Qs[r][c4 * 4]);
    }
    for (int e = tid; e < 64 * 64; e += 128) {     // e_r transposed (manual)
        int r = e >> 6, c = e & 63;
        Es[c][r] = e_r[(size_t)(kb + r) * HDIM + c];
    }
    wait_async0();
    __syncthreads();

    v8f acc[4];
    #pragma unroll
    for (int i = 0; i < 4; ++i) acc[i] = v8f_zero();

    #pragma unroll
    for (int d = 0; d < HDIM; d += 4) {
        v2f a;
        a.x = Qs[16 * wv + lm][d + 2 * half];
        a.y = Qs[16 * wv + lm][d + 2 * half + 1];
        #pragma unroll
        for (int nf = 0; nf < 4; ++nf) {
            v2f b;
            b.x = Es[d + 2 * half][nf * 16 + lm];
            b.y = Es[d + 2 * half + 1][nf * 16 + lm];
            acc[nf] = wmma4(a, b, acc[nf]);
        }
    }

    #pragma unroll
    for (int nf = 0; nf < 4; ++nf) {
        const int kp = kb + nf * 16 + lm;
        #pragma unroll
        for (int i = 0; i < 8; ++i) {
            const int q = qb + 16 * wv + 8 * half + i;
            const float msk = (kp >= (S_DIM - 1 - q)) ? 1.0f : 0.0f;
            qe[((size_t)bh << 20) + ((size_t)q << 10) + kp] = acc[nf][i] * msk;
        }
    }
}

// ---------------------------------------------------------------------------
// S_rel = skew(qe): f=(i+1)*1024+j; r=f/1025; c=f%1025; (c==0)?0:qe[r][c-1]
// 4 elements per thread, float4 store (pure-bandwidth kernel).
// ---------------------------------------------------------------------------
__global__ void skew_kernel(const float* __restrict__ qe, float* __restrict__ S_rel) {
    const size_t i4 = ((size_t)blockIdx.x * 256 + threadIdx.x) * 4;
    const int bh  = (int)(i4 >> 20);
    const int rem = (int)(i4 & 0xFFFFF);
    const int i = rem >> 10, j = rem & 1023;
    const float* src = qe + ((size_t)bh << 20);
    float4 o;
    float* op = &o.x;
    #pragma unroll
    for (int jj = 0; jj < 4; ++jj) {
        const int f = (i + 1) * 1024 + j + jj;
        const int r = f / 1025, c = f % 1025;
        op[jj] = (c == 0) ? 0.0f : src[((size_t)r << 10) + (c - 1)];
    }
    *(float4*)(S_rel + i4) = o;
}

// ---------------------------------------------------------------------------
// Attention, two-pass flash softmax. One 16-row q-tile per wave.
// K^T and V tiles are wave-private, double-buffered, async-staged LDS tiles:
// both passes are ds_load -> v_wmma with next tile's DMA in flight.
// ---------------------------------------------------------------------------
__global__ __launch_bounds__(128)
void attn_kernel(const float* __restrict__ qh, const float* __restrict__ khT,
                 const float* __restrict__ vh, const float* __restrict__ S_rel,
                 float* __restrict__ out) {
    __shared__ float Qs[4][16][68];
    __shared__ float Pt[4][16][20];
    __shared__ float Ks[4][2][64][20];   // K^T tile: [d][k_local]
    __shared__ float Vs[4][2][16][72];   // V tile:   [k_local][d]

    const int tid  = threadIdx.x;
    const int lane = tid & 31;
    const int wv   = tid >> 5;
    const int half = lane >> 4;
    const int lm   = lane & 15;

    const int g  = blockIdx.x * 4 + wv;
    const int bh = g >> 6;
    const int qt = g & 63;
    const int qb = qt * 16;
    const int b  = bh >> 4, h = bh & 15;

    const float* Q  = qh  + (size_t)bh * S_DIM * HDIM;
    const float* KT = khT + (size_t)bh * S_DIM * HDIM;   // [D][S]
    const float* V  = vh  + (size_t)bh * S_DIM * HDIM;   // [S][D]
    const float* SR = S_rel + ((size_t)bh << 20);

    auto stageK = [&](int t, int bb) {
        #pragma unroll
        for (int i = 0; i < 8; ++i) {                // 64x16 tile
            int e4 = i * 32 + lane;
            int d = e4 >> 2, c4 = e4 & 3;
            async_copy16(KT + (size_t)d * S_DIM + t * 16 + c4 * 4,
                         &Ks[wv][bb][d][c4 * 4]);
        }
    };
    auto stageV = [&](int t, int bb) {
        #pragma unroll
        for (int i = 0; i < 8; ++i) {                // 16x64 tile
            int e4 = i * 32 + lane;
            int r = e4 >> 4, c4 = e4 & 15;
            async_copy16(V + (size_t)(t * 16 + r) * HDIM + c4 * 4,
                         &Vs[wv][bb][r][c4 * 4]);
        }
    };

    #pragma unroll
    for (int i = 0; i < 8; ++i) {                    // Q tile, async b128
        int e4 = i * 32 + lane;
        int r = e4 >> 4, c4 = e4 & 15;
        async_copy16(Q + (size_t)(qb + r) * HDIM + c4 * 4, &Qs[wv][r][c4 * 4]);
    }

    float m[8], ssum[8];
    #pragma unroll
    for (int i = 0; i < 8; ++i) { m[i] = -3.0e38f; ssum[i] = 0.0f; }

    // ---- Pass 1: online row max / sum ----
    stageK(0, 0);
    int buf = 0;
    for (int t = 0; t <= qt; ++t) {
        wait_async0();
        wait_ds0();
        if (t < qt) stageK(t + 1, buf ^ 1);

        v8f sc = v8f_zero();
        #pragma unroll
        for (int d = 0; d < HDIM; d += 4) {
            v2f a;
            a.x = Qs[wv][lm][d + 2 * half];
            a.y = Qs[wv][lm][d + 2 * half + 1];
            v2f bf;
            bf.x = Ks[wv][buf][d + 2 * half][lm];
            bf.y = Ks[wv][buf][d + 2 * half + 1][lm];
            sc = wmma4(a, bf, sc);
        }
        const int kp = t * 16 + lm;
        #pragma unroll
        for (int i = 0; i < 8; ++i) {
            const int qrow = qb + 8 * half + i;
            const float srel = SR[((size_t)qrow << 10) + kp];
            const float val = ((kp <= qrow) ? sc[i] * 0.125f : -1.0e9f) + srel;
            const float mn = fmaxf(m[i], val);
            ssum[i] = ssum[i] * __expf(m[i] - mn) + __expf(val - mn);
            m[i] = mn;
        }
        buf ^= 1;
    }
    // Cross-lane reduce over the 16 column-lanes (halves independent).
    #pragma unroll
    for (int off = 1; off < 16; off <<= 1) {
        #pragma unroll
        for (int i = 0; i < 8; ++i) {
            const float om = __shfl_xor(m[i], off, 32);
            const float os = __shfl_xor(ssum[i], off, 32);
            const float mn = fmaxf(m[i], om);
            ssum[i] = ssum[i] * __expf(m[i] - mn) + os * __expf(om - mn);
            m[i] = mn;
        }
    }
    float rinv[8];
    #pragma unroll
    for (int i = 0; i < 8; ++i) rinv[i] = 1.0f / ssum[i];

    // ---- Pass 2: P = exp(score - max); O += P.V ----
    v8f o[4];
    #pragma unroll
    for (int i = 0; i < 4; ++i) o[i] = v8f_zero();

    wait_ds0();                       // pass-1 LDS reads done before restage
    stageK(0, 0);
    stageV(0, 0);
    buf = 0;
    for (int t = 0; t <= qt; ++t) {
        wait_async0();
        wait_ds0();
        if (t < qt) { stageK(t + 1, buf ^ 1); stageV(t + 1, buf ^ 1); }

        v8f sc = v8f_zero();
        #pragma unroll
        for (int d = 0; d < HDIM; d += 4) {
            v2f a;
            a.x = Qs[wv][lm][d + 2 * half];
            a.y = Qs[wv][lm][d + 2 * half + 1];
            v2f bf;
            bf.x = Ks[wv][buf][d + 2 * half][lm];
            bf.y = Ks[wv][buf][d + 2 * half + 1][lm];
            sc = wmma4(a, bf, sc);
        }
        const int kp = t * 16 + lm;
        #pragma unroll
        for (int i = 0; i < 8; ++i) {
            const int qrow = qb + 8 * half + i;
            const float srel = SR[((size_t)qrow << 10) + kp];
            const float val = ((kp <= qrow) ? sc[i] * 0.125f : -1.0e9f) + srel;
            Pt[wv][8 * half + i][lm] = __expf(val - m[i]);
        }
        wait_ds0();                   // C-layout -> A-layout via wave LDS

        #pragma unroll
        for (int kk = 0; kk < 16; kk += 4) {
            v2f a;
            a.x = Pt[wv][lm][kk + 2 * half];
            a.y = Pt[wv][lm][kk + 2 * half + 1];
            #pragma unroll
            for (int nf = 0; nf < 4; ++nf) {
                v2f bf;
                bf.x = Vs[wv][buf][kk + 2 * half][nf * 16 + lm];
                bf.y = Vs[wv][buf][kk + 2 * half + 1][nf * 16 + lm];
                o[nf] = wmma4(a, bf, o[nf]);
            }
        }
        buf ^= 1;
    }

    #pragma unroll
    for (int nf = 0; nf < 4; ++nf) {
        #pragma unroll
        for (int i = 0; i < 8; ++i) {
            const int s = qb + 8 * half + i;
            const int d = nf * 16 + lm;
            out[((size_t)b * S_DIM + s) * D_MODEL + h * HDIM + d] = o[nf][i] * rinv[i];
        }
    }
}

// ---------------------------------------------------------------------------
extern "C" void kernel_launch(void* const* d_in, const int* in_sizes, int n_in,
                              void* d_out, int out_size, void* d_ws, size_t ws_size,
                              hipStream_t stream) {
    (void)in_sizes; (void)n_in; (void)out_size; (void)ws_size;

    const float* q   = (const float*)d_in[0];
    const float* k   = (const float*)d_in[1];
    const float* v   = (const float*)d_in[2];
    /* d_in[3] = mask: analytic causal tril, not read */
    const float* Wq  = (const float*)d_in[4];
    const float* bq  = (const float*)d_in[5];
    const float* Wk  = (const float*)d_in[6];
    const float* bk  = (const float*)d_in[7];
    const float* Wv  = (const float*)d_in[8];
    const float* bv  = (const float*)d_in[9];
    const float* Wo  = (const float*)d_in[10];
    const float* bo  = (const float*)d_in[11];
    const float* e_r = (const float*)d_in[12];

    const size_t HEADS_ELEMS = (size_t)B_DIM * NHEAD * S_DIM * HDIM;
    float* ws   = (float*)d_ws;
    float* qh   = ws;
    float* khT  = ws + HEADS_ELEMS;
    float* vh   = ws + 2 * HEADS_ELEMS;
    float* attn = ws + 3 * HEADS_ELEMS;

    const size_t OUT_ELEMS  = (size_t)B_DIM * S_DIM * D_MODEL;
    const size_t SREL_ELEMS = (size_t)B_DIM * NHEAD * S_DIM * S_DIM;
    float* out_p  = (float*)d_out;
    float* srel_p = out_p + OUT_ELEMS;
    float* qe_p   = srel_p + SREL_ELEMS;

    dim3 ggemm(NROWS / 64, D_MODEL / 64);
    gemm_proj<<<ggemm, 128, 0, stream>>>(q, Wq, bq, qh, 0);
    gemm_proj<<<ggemm, 128, 0, stream>>>(k, Wk, bk, khT, 1);
    gemm_proj<<<ggemm, 128, 0, stream>>>(v, Wv, bv, vh, 2);

    dim3 gqe(S_DIM / 64, S_DIM / 64, B_DIM * NHEAD);
    qe_kernel<<<gqe, 128, 0, stream>>>(qh, e_r, qe_p);

    skew_kernel<<<(unsigned)(SREL_ELEMS / 4 / 256), 256, 0, stream>>>(qe_p, srel_p);

    attn_kernel<<<(B_DIM * NHEAD * (S_DIM / 16)) / 4, 128, 0, stream>>>(
        qh, khT, vh, srel_p, attn);

    gemm_proj<<<ggemm, 128, 0, stream>>>(attn, Wo, bo, out_p, 3);
}